// ReferenceProtoHead_62113817035466
// MI455X (gfx1250) — compile-verified
//
#include <hip/hip_runtime.h>
#include <hip/hip_bf16.h>

// ---------------------------------------------------------------------------
// ProtoHead for MI455X (gfx1250, wave32, WMMA)
//   prototypes = segment_mean(z_support, y_support)          [C, D]
//   out[q,c]   = -(||q||^2 + ||p_c||^2 - 2 q.p_c)            [Q, C]
// Output-bandwidth bound (512MB out @ 23.3TB/s ~ 22us); f32 WMMA keeps
// reference precision at zero perf cost. Prototype tile staged into LDS
// via CDNA5 async-to-LDS DMA (ASYNCcnt) instead of a VGPR round trip.
// ---------------------------------------------------------------------------

#define D_DIM 256
#define C_DIM 1024
#define LDS_STRIDE (D_DIM + 4)   // 260: 260 % 64 == 4 -> conflict-free ds_load_b64
#define WAVES_PER_BLOCK 8
#define TILE_Q (16 * WAVES_PER_BLOCK)  // 128 query rows per block
#define TILE_C 16                      // 16 classes per block

typedef float v2f __attribute__((ext_vector_type(2)));
typedef float v8f __attribute__((ext_vector_type(8)));

// ---- kernel 1: zero workspace accumulators --------------------------------
__global__ void ph_zero_kernel(float* __restrict__ p, int n) {
    int i = blockIdx.x * blockDim.x + threadIdx.x;
    if (i < n) p[i] = 0.0f;
}

// ---- kernel 2: scatter-add class sums + counts ----------------------------
__global__ __launch_bounds__(D_DIM)
void ph_accum_kernel(const float* __restrict__ zs, const int* __restrict__ ys,
                     float* __restrict__ sums, float* __restrict__ counts, int ns) {
    const int d = threadIdx.x;  // 0..255 == one feature column
    for (int row = blockIdx.x; row < ns; row += gridDim.x) {
        const int label = ys[row];
        atomicAdd(&sums[(size_t)label * D_DIM + d], zs[(size_t)row * D_DIM + d]);
        if (d == 0) atomicAdd(&counts[label], 1.0f);
    }
}

// ---- kernel 3: prototypes = sums / counts ; p2[c] = ||p_c||^2 -------------
__global__ __launch_bounds__(D_DIM)
void ph_finalize_kernel(float* __restrict__ proto, const float* __restrict__ counts,
                        float* __restrict__ p2) {
    __shared__ float red[D_DIM];
    const int c = blockIdx.x;
    const int d = threadIdx.x;
    const float inv = 1.0f / fmaxf(counts[c], 1.0f);
    const float v = proto[(size_t)c * D_DIM + d] * inv;
    proto[(size_t)c * D_DIM + d] = v;   // divide in place (each element touched once)
    red[d] = v * v;
    __syncthreads();
    #pragma unroll
    for (int s = D_DIM / 2; s > 0; s >>= 1) {
        if (d < s) red[d] += red[d + s];
        __syncthreads();
    }
    if (d == 0) p2[c] = red[0];
}

// ---- kernel 4: WMMA GEMM epilogue-fused distance --------------------------
// Block: 256 threads = 8 waves. Wave w computes 16x16 tile at
// rows [blockIdx.x*128 + 16w, +16), cols [blockIdx.y*16, +16).
// Prototype tile (16 x 256 f32, padded) staged once in LDS via async DMA,
// shared by all 8 waves.
__global__ __launch_bounds__(256)
void ph_gemm_kernel(const float* __restrict__ zq, const float* __restrict__ proto,
                    const float* __restrict__ p2, float* __restrict__ out, int q_total) {
    __shared__ float sB[TILE_C * LDS_STRIDE];  // 16*260*4 = 16640 B

    const int tid   = threadIdx.x;
    const int wave  = tid >> 5;
    const int lane  = tid & 31;
    const int cbase = blockIdx.y * TILE_C;
    const int qbase = blockIdx.x * TILE_Q + wave * 16;

    // Stage prototype tile into LDS with CDNA5 async global->LDS DMA
    // (GLOBAL_LOAD_ASYNC_TO_LDS_B128, tracked by ASYNCcnt; no VGPR round trip).
    // 16 rows x 64 float4 = 1024 b128 transfers; 4 per thread, EXEC all-ones.
    #pragma unroll
    for (int i = 0; i < 4; ++i) {
        const int idx  = tid + i * 256;          // 0..1023
        const int row  = idx >> 6;               // /64 float4s per row
        const int col4 = idx & 63;
        const float* src = proto + (size_t)(cbase + row) * D_DIM + col4 * 4;
        // Low 32 bits of a generic shared-memory address == LDS byte offset.
        const unsigned lds_off =
            (unsigned)(uintptr_t)&sB[row * LDS_STRIDE + col4 * 4];
        asm volatile("global_load_async_to_lds_b128 %0, %1, off"
                     :: "v"(lds_off), "v"(src) : "memory");
    }
    asm volatile("s_wait_asynccnt 0x0" ::: "memory");
    __syncthreads();

    // WMMA fragment addressing (ISA 7.12.2, f32 16x16x4):
    //   A: lanes 0-15 hold rows m=0..15 at K={k,k+1}; lanes 16-31 at K={k+2,k+3}
    //   B: symmetric over columns n=0..15 (K-major).
    const int m    = lane & 15;
    const int koff = (lane >> 4) << 1;  // 0 or 2
    const float* aRow = zq + (size_t)(qbase + m) * D_DIM + koff;
    const float* bRow = &sB[m * LDS_STRIDE + koff];

    v8f acc = {};
    float sq = 0.0f;  // on-the-fly ||q||^2 partial for row m, this half's K slots
    #pragma unroll 4
    for (int k = 0; k < D_DIM; k += 4) {
        const v2f a = *(const v2f*)(aRow + k);  // global_load_b64 (s_clause'd)
        const v2f b = *(const v2f*)(bRow + k);  // ds_load_2addr_b64 (conflict-free)
        sq += a.x * a.x + a.y * a.y;
        acc = __builtin_amdgcn_wmma_f32_16x16x4_f32(
            /*neg_a=*/false, a, /*neg_b=*/false, b,
            /*c_mod=*/(short)0, acc, /*reuse_a=*/false, /*reuse_b=*/false);
    }

    // Combine the two half-wave K partials: lane j now holds q2 of row (j&15).
    const float q2m = sq + __shfl_xor(sq, 16, 32);
    const float p2n = p2[cbase + m];            // this lane's output column
    const int   col = cbase + m;
    const int rowhi = (lane >> 4) * 8;          // C/D layout: hi half -> rows +8

    #pragma unroll
    for (int r = 0; r < 8; ++r) {
        const float q2r = __shfl(q2m, r + rowhi, 32);  // ds_bpermute broadcast
        const int   row = qbase + r + rowhi;
        out[(size_t)row * C_DIM + col] = 2.0f * acc[r] - q2r - p2n;
    }
    (void)q_total;
}

// ---------------------------------------------------------------------------
extern "C" void kernel_launch(void* const* d_in, const int* in_sizes, int n_in,
                              void* d_out, int out_size, void* d_ws, size_t ws_size,
                              hipStream_t stream) {
    const float* z_support = (const float*)d_in[0];
    const int*   y_support = (const int*)d_in[1];
    const float* z_query   = (const float*)d_in[2];
    float*       out       = (float*)d_out;

    // Workspace: [C*D sums->prototypes][C counts][C p2]
    float* sums   = (float*)d_ws;
    float* counts = sums + (size_t)C_DIM * D_DIM;
    float* p2     = counts + C_DIM;

    const int ns = in_sizes[0] / D_DIM;
    const int q  = in_sizes[2] / D_DIM;

    const int n_zero = C_DIM * D_DIM + C_DIM;
    ph_zero_kernel<<<(n_zero + 255) / 256, 256, 0, stream>>>(sums, n_zero);

    ph_accum_kernel<<<4096, D_DIM, 0, stream>>>(z_support, y_support, sums, counts, ns);

    ph_finalize_kernel<<<C_DIM, D_DIM, 0, stream>>>(sums, counts, p2);

    dim3 grid(q / TILE_Q, C_DIM / TILE_C);  // 1024 x 64 blocks
    ph_gemm_kernel<<<grid, 256, 0, stream>>>(z_query, sums, p2, out, q);

    (void)n_in; (void)out_size; (void)ws_size;
}